// HeightMerging_25443386261905
// MI455X (gfx1250) — compile-verified
//
#include <hip/hip_runtime.h>

#define BB   4
#define HH   176
#define WW   200
#define CIN  128
#define COUT 256
#define BWIN 4096
#define NPW  64
#define NPTS (BWIN * NPW)          /* 262144 points */
#define NVOX (BB * WW * HH)        /* 140800 voxels */
#define EPSV 1e-5f
#define A_STRIDE 132               /* 16x4B pad -> conflict-free ds_load_b64, 16B aligned rows */

typedef __attribute__((ext_vector_type(2))) float v2f;
typedef __attribute__((ext_vector_type(4))) float v4f;
typedef __attribute__((ext_vector_type(8))) float v8f;

// ---------------------------------------------------------------- zero dense
__global__ void k_zero(v4f* __restrict__ p, int n4) {
    int i = blockIdx.x * blockDim.x + threadIdx.x;
    if (i < n4) p[i] = (v4f){0.f, 0.f, 0.f, 0.f};
}

// ------------------------------------------------- scatter-add points->dense
// one wave per point: 32 lanes x 4 channels = 128 channels
__global__ __launch_bounds__(256) void k_scatter(const float* __restrict__ feats,
                                                 const int*   __restrict__ idx,
                                                 float*       __restrict__ dense) {
    int p    = blockIdx.x * 8 + (threadIdx.x >> 5);
    int lane = threadIdx.x & 31;
    if (p >= NPTS) return;
    int b = idx[p * 3 + 0];
    int x = idx[p * 3 + 1];
    int y = idx[p * 3 + 2];
    if (x < 0 || x >= HH || y < 0 || y >= WW) return;   // dropped, like ref
    size_t vox = ((size_t)b * WW + y) * HH + x;
    const v4f f = __builtin_nontemporal_load(
        (const v4f*)(feats + (size_t)p * CIN + lane * 4));
    float* dst = dense + vox * CIN + lane * 4;
    __hip_atomic_fetch_add(dst + 0, f.x, __ATOMIC_RELAXED, __HIP_MEMORY_SCOPE_AGENT);
    __hip_atomic_fetch_add(dst + 1, f.y, __ATOMIC_RELAXED, __HIP_MEMORY_SCOPE_AGENT);
    __hip_atomic_fetch_add(dst + 2, f.z, __ATOMIC_RELAXED, __HIP_MEMORY_SCOPE_AGENT);
    __hip_atomic_fetch_add(dst + 3, f.w, __ATOMIC_RELAXED, __HIP_MEMORY_SCOPE_AGENT);
}

// -------------------------------------------------- per-voxel layernorm (in place)
// one wave per voxel; wave32 shuffle reduction
__global__ __launch_bounds__(256) void k_norm(float* __restrict__ dense,
                                              const float* __restrict__ nw,
                                              const float* __restrict__ nb) {
    int v    = blockIdx.x * 8 + (threadIdx.x >> 5);
    int lane = threadIdx.x & 31;
    if (v >= NVOX) return;
    float* row = dense + (size_t)v * CIN + lane * 4;
    v4f f = *(const v4f*)row;
    float s = f.x + f.y + f.z + f.w;
    float q = f.x * f.x + f.y * f.y + f.z * f.z + f.w * f.w;
#pragma unroll
    for (int off = 16; off > 0; off >>= 1) {
        s += __shfl_xor(s, off, 32);
        q += __shfl_xor(q, off, 32);
    }
    float mu  = s * (1.0f / CIN);
    float var = q * (1.0f / CIN) - mu * mu;
    float r   = rsqrtf(var + EPSV);
    v4f w4 = *(const v4f*)(nw + lane * 4);
    v4f b4 = *(const v4f*)(nb + lane * 4);
    f = (f - mu) * r * w4 + b4;
    *(v4f*)row = f;
}

// ---------------- fused gather + (points x 128) @ (128 x 256) GEMM via WMMA f32
// block = 128 threads (4 waves) = 16 points x 256 outputs
// wave w covers N in [w*64, w*64+64) as four 16x16 tiles, K loop 128/4
__global__ __launch_bounds__(128) void k_gemm(const float* __restrict__ dense,
                                              const int*   __restrict__ idx,
                                              const float* __restrict__ lw,
                                              float*       __restrict__ out) {
    __shared__ float As[16 * A_STRIDE];
    const int t     = threadIdx.x;
    const int pbase = blockIdx.x * 16;

    // stage 16 normalized point-rows (or zeros for invalid) into LDS
    {
        int pl  = t >> 3;       // point 0..15
        int seg = t & 7;        // 16-float segment 0..7
        int p   = pbase + pl;
        int b = idx[p * 3 + 0];
        int x = idx[p * 3 + 1];
        int y = idx[p * 3 + 2];
        bool valid = (x >= 0) & (x < HH) & (y >= 0) & (y < WW);
        const float* src =
            dense + (((size_t)(valid ? b : 0) * WW + (valid ? y : 0)) * HH +
                     (valid ? x : 0)) * CIN + seg * 16;
        float* dst = As + pl * A_STRIDE + seg * 16;
#pragma unroll
        for (int i = 0; i < 16; i += 4) {
            v4f f = valid ? *(const v4f*)(src + i) : (v4f){0.f, 0.f, 0.f, 0.f};
            *(v4f*)(dst + i) = f;   // (pl*132 + seg*16 + i)*4 is 16B aligned
        }
    }
    __syncthreads();

    const int wave = t >> 5;
    const int lane = t & 31;
    const int ln   = lane & 15;   // A: M row / B,C: N column
    const int hi   = lane >> 4;   // selects K pair (A/B) and M half (C/D)
    const int nb0  = wave * 64;

    v8f acc0 = {}; v8f acc1 = {}; v8f acc2 = {}; v8f acc3 = {};
    const float* asrc = As + ln * A_STRIDE + 2 * hi;
    const float* lrow = lw + (size_t)(nb0 + ln) * CIN + 2 * hi;

#pragma unroll 4
    for (int k0 = 0; k0 < CIN; k0 += 4) {
        v2f a  = *(const v2f*)(asrc + k0);                  // ds_load_b64
        v2f b0 = *(const v2f*)(lrow + k0);                  // L2-resident weights
        v2f b1 = *(const v2f*)(lrow + k0 + 16 * CIN);
        v2f b2 = *(const v2f*)(lrow + k0 + 32 * CIN);
        v2f b3 = *(const v2f*)(lrow + k0 + 48 * CIN);
        acc0 = __builtin_amdgcn_wmma_f32_16x16x4_f32(false, a, false, b0, (short)0, acc0, false, false);
        acc1 = __builtin_amdgcn_wmma_f32_16x16x4_f32(false, a, false, b1, (short)0, acc1, false, false);
        acc2 = __builtin_amdgcn_wmma_f32_16x16x4_f32(false, a, false, b2, (short)0, acc2, false, false);
        acc3 = __builtin_amdgcn_wmma_f32_16x16x4_f32(false, a, false, b3, (short)0, acc3, false, false);
    }

    // C/D layout: element e -> (M = e + 8*hi, N = ln); write-once -> NT stores
    float* obase = out + (size_t)pbase * COUT + nb0 + ln;
#pragma unroll
    for (int e = 0; e < 8; ++e) {
        float* orow = obase + (size_t)(e + 8 * hi) * COUT;
        __builtin_nontemporal_store(acc0[e], orow);
        __builtin_nontemporal_store(acc1[e], orow + 16);
        __builtin_nontemporal_store(acc2[e], orow + 32);
        __builtin_nontemporal_store(acc3[e], orow + 48);
    }
}

// ----------------------------------------------------------------------------
extern "C" void kernel_launch(void* const* d_in, const int* in_sizes, int n_in,
                              void* d_out, int out_size, void* d_ws, size_t ws_size,
                              hipStream_t stream) {
    const float* feats = (const float*)d_in[0];
    const int*   idx   = (const int*)d_in[1];
    const float* nw    = (const float*)d_in[2];
    const float* nbias = (const float*)d_in[3];
    const float* lw    = (const float*)d_in[4];
    float* out   = (float*)d_out;
    float* dense = (float*)d_ws;                 // NVOX*CIN f32 = ~72 MB

    const int n4 = NVOX * CIN / 4;               // 4,505,600 float4
    k_zero   <<<(n4 + 255) / 256, 256, 0, stream>>>((v4f*)dense, n4);
    k_scatter<<<NPTS / 8,        256, 0, stream>>>(feats, idx, dense);
    k_norm   <<<NVOX / 8,        256, 0, stream>>>(dense, nw, nbias);
    k_gemm   <<<NPTS / 16,       128, 0, stream>>>(dense, idx, lw, out);

    // second tuple output: window_indices bit-copied to the tail of d_out
    (void)hipMemcpyAsync(out + (size_t)NPTS * COUT, idx,
                         (size_t)NPTS * 3 * sizeof(int),
                         hipMemcpyDeviceToDevice, stream);
}